// GlobalEncoder_62758062129335
// MI455X (gfx1250) — compile-verified
//
#include <hip/hip_runtime.h>
#include <math.h>

typedef __attribute__((ext_vector_type(2))) float v2f;
typedef __attribute__((ext_vector_type(8))) float v8f;

#define DFEAT 64

// ---------------------------------------------------------------------------
// Per-layer init: out[n,d] = bias[d]; m[n] = -inf; denom[n] = 0
// ---------------------------------------------------------------------------
__global__ __launch_bounds__(256) void init_nodes(float* __restrict__ out,
                                                  float* __restrict__ m,
                                                  float* __restrict__ denom,
                                                  const float* __restrict__ bias,
                                                  int n) {
  int i = blockIdx.x * blockDim.x + threadIdx.x;
  if (i < n * DFEAT) out[i] = bias[i & (DFEAT - 1)];
  if (i < n) { m[i] = -INFINITY; denom[i] = 0.0f; }
}

// ---------------------------------------------------------------------------
// h = x @ W  (fp32 WMMA 16x16x4), fused a_src = h@att_src, a_dst = h@att_dst.
// One wave owns a 16(row) x 64(col) tile of h: 4 accumulators, 16 K-steps,
// 64 v_wmma_f32_16x16x4_f32 per wave.
// W is staged in LDS K-pair-interleaved: sW2[p][c] = {W[2p][c], W[2p+1][c]},
// so each B fragment is one aligned ds_load_b64 straight into a VGPR pair
// (no v_mov packing in the WMMA inner loop).
// ---------------------------------------------------------------------------
__global__ __launch_bounds__(256) void gemm_proj(const float* __restrict__ x,
                                                 const float* __restrict__ W,
                                                 const float* __restrict__ att_src,
                                                 const float* __restrict__ att_dst,
                                                 float* __restrict__ h,
                                                 float* __restrict__ asrc,
                                                 float* __restrict__ adst,
                                                 int n) {
  __shared__ v2f  sW2[(DFEAT / 2) * DFEAT];   // 16 KB, pair-interleaved W
  __shared__ float sAs[DFEAT];
  __shared__ float sAd[DFEAT];

  // Permuting store: W[r][c] -> sW2[r/2][c] component r&1
  for (int i = threadIdx.x; i < DFEAT * DFEAT; i += 256) {
    const int r = i >> 6;
    const int c = i & 63;
    ((float*)sW2)[((r >> 1) * DFEAT + c) * 2 + (r & 1)] = W[i];
  }
  if (threadIdx.x < DFEAT) {
    sAs[threadIdx.x] = att_src[threadIdx.x];
    sAd[threadIdx.x] = att_dst[threadIdx.x];
  }
  __syncthreads();

  const int wave = threadIdx.x >> 5;   // 8 waves / block
  const int lane = threadIdx.x & 31;
  const int half = lane >> 4;          // 0: lanes 0-15, 1: lanes 16-31
  const int l16  = lane & 15;
  const int rowBase = (blockIdx.x * 8 + wave) * 16;

  // A fragment (16x4 f32 layout): lane L holds row = L%16,
  // VGPR0 -> K = half*2, VGPR1 -> K = half*2 + 1 within each K-step of 4.
  int arow  = rowBase + l16;
  int arowc = arow < n ? arow : (n - 1);           // clamp, keep EXEC full
  const float* __restrict__ xrow = x + (size_t)arowc * DFEAT;

  v8f acc0 = {}, acc1 = {}, acc2 = {}, acc3 = {};

#pragma unroll
  for (int ks = 0; ks < 16; ++ks) {
    const int k = ks * 4 + half * 2;   // this lane's K-pair base
    const int p = k >> 1;              // pair index into sW2

    v2f a;
    a.x = xrow[k];
    a.y = xrow[k + 1];

    // B fragments (4x16 f32 layout): col = L%16 (+tile*16); one b64 each.
    const v2f* __restrict__ bp = &sW2[p * DFEAT + l16];
    const v2f b0 = bp[0];
    const v2f b1 = bp[16];
    const v2f b2 = bp[32];
    const v2f b3 = bp[48];

    acc0 = __builtin_amdgcn_wmma_f32_16x16x4_f32(false, a, false, b0, (short)0, acc0, false, false);
    acc1 = __builtin_amdgcn_wmma_f32_16x16x4_f32(false, a, false, b1, (short)0, acc1, false, false);
    acc2 = __builtin_amdgcn_wmma_f32_16x16x4_f32(false, a, false, b2, (short)0, acc2, false, false);
    acc3 = __builtin_amdgcn_wmma_f32_16x16x4_f32(false, a, false, b3, (short)0, acc3, false, false);
  }

  // C/D layout: VGPR j, lanes 0-15 -> row M=j, lanes 16-31 -> row M=j+8,
  // N = lane%16 within each 16-column tile.
#pragma unroll
  for (int j = 0; j < 8; ++j) {
    const int row = rowBase + j + half * 8;
    if (row < n) {
      float* __restrict__ hr = h + (size_t)row * DFEAT;
      hr[l16 +  0] = acc0[j];
      hr[l16 + 16] = acc1[j];
      hr[l16 + 32] = acc2[j];
      hr[l16 + 48] = acc3[j];
    }
  }

  // Fused attention logits: reduce partial dots across each 16-lane half
  // (xor masks 8/4/2/1 never cross bit 4, so halves stay independent).
#pragma unroll
  for (int j = 0; j < 8; ++j) {
    float ps = acc0[j] * sAs[l16] + acc1[j] * sAs[l16 + 16] +
               acc2[j] * sAs[l16 + 32] + acc3[j] * sAs[l16 + 48];
    float pd = acc0[j] * sAd[l16] + acc1[j] * sAd[l16 + 16] +
               acc2[j] * sAd[l16 + 32] + acc3[j] * sAd[l16 + 48];
#pragma unroll
    for (int off = 8; off > 0; off >>= 1) {
      ps += __shfl_xor(ps, off, 32);
      pd += __shfl_xor(pd, off, 32);
    }
    const int row = rowBase + j + half * 8;
    if (l16 == 0 && row < n) {
      asrc[row] = ps;
      adst[row] = pd;
    }
  }
}

// ---------------------------------------------------------------------------
// Sign-aware float atomic max via integer atomics (IEEE total order trick).
// ---------------------------------------------------------------------------
__device__ __forceinline__ void atomicMaxFloat(float* addr, float v) {
  if (v >= 0.0f)
    atomicMax((int*)addr, __float_as_int(v));
  else
    atomicMin((unsigned int*)addr, __float_as_uint(v));
}

__global__ __launch_bounds__(256) void edge_logits_max(const int* __restrict__ src,
                                                       const int* __restrict__ dst,
                                                       const float* __restrict__ as,
                                                       const float* __restrict__ ad,
                                                       float* __restrict__ ebuf,
                                                       float* __restrict__ m,
                                                       int E) {
  int i = blockIdx.x * blockDim.x + threadIdx.x;
  if (i >= E) return;
  const int s = src[i];
  const int d = dst[i];
  float e = as[s] + ad[d];
  e = e > 0.0f ? e : 0.2f * e;   // leaky_relu, slope 0.2
  ebuf[i] = e;
  atomicMaxFloat(&m[d], e);
}

__global__ __launch_bounds__(256) void edge_exp(const int* __restrict__ dst,
                                                float* __restrict__ ebuf,
                                                const float* __restrict__ m,
                                                float* __restrict__ denom,
                                                int E) {
  int i = blockIdx.x * blockDim.x + threadIdx.x;
  if (i >= E) return;
  const int d = dst[i];
  const float ex = __expf(ebuf[i] - m[d]);
  ebuf[i] = ex;
  atomicAdd(&denom[d], ex);
}

// One wave per edge: lane l moves h[src][2l..2l+1] (coalesced 256B gather)
// and scatter-adds alpha * h[src] into out[dst] (L2-resident atomics).
__global__ __launch_bounds__(256) void edge_scatter(const int* __restrict__ src,
                                                    const int* __restrict__ dst,
                                                    const float* __restrict__ ebuf,
                                                    const float* __restrict__ denom,
                                                    const float* __restrict__ h,
                                                    float* __restrict__ out,
                                                    int E) {
  const int e = (int)((blockIdx.x * (size_t)blockDim.x + threadIdx.x) >> 5);
  const int lane = threadIdx.x & 31;
  if (e >= E) return;
  const int s = src[e];
  const int d = dst[e];
  const float w = ebuf[e] / (denom[d] + 1e-16f);
  const float2 v = ((const float2*)(h + (size_t)s * DFEAT))[lane];
  float* op = out + (size_t)d * DFEAT + lane * 2;
  atomicAdd(op + 0, w * v.x);
  atomicAdd(op + 1, w * v.y);
}

// ---------------------------------------------------------------------------
// Driver: 4 layers, ping-pong node features through workspace, last -> d_out.
// ---------------------------------------------------------------------------
extern "C" void kernel_launch(void* const* d_in, const int* in_sizes, int n_in,
                              void* d_out, int out_size, void* d_ws, size_t ws_size,
                              hipStream_t stream) {
  const float* g        = (const float*)d_in[0];
  const int*   edge_idx = (const int*)d_in[1];   // [2, E]
  const float* W        = (const float*)d_in[2];
  const float* att_src  = (const float*)d_in[3];
  const float* att_dst  = (const float*)d_in[4];
  const float* bias     = (const float*)d_in[5];

  const int N = in_sizes[0] / DFEAT;
  const int E = in_sizes[1] / 2;
  const int* srcI = edge_idx;
  const int* dstI = edge_idx + E;

  float* ws = (float*)d_ws;
  float* h  = ws;                         // N*64
  float* xA = h  + (size_t)N * DFEAT;     // N*64
  float* xB = xA + (size_t)N * DFEAT;     // N*64
  float* as = xB + (size_t)N * DFEAT;     // N
  float* ad = as + N;                     // N
  float* m  = ad + N;                     // N
  float* dn = m  + N;                     // N
  float* eb = dn + N;                     // E

  const int nodeBlocks = (N * DFEAT + 255) / 256;
  const int gemmBlocks = (N + 127) / 128;             // 8 waves x 16 rows
  const int edgeBlocks = (E + 255) / 256;
  const int scatBlocks = (int)(((size_t)E * 32 + 255) / 256);

  const float* xin = g;
  for (int layer = 0; layer < 4; ++layer) {
    float* xout = (layer == 3) ? (float*)d_out : ((layer & 1) ? xB : xA);

    init_nodes<<<nodeBlocks, 256, 0, stream>>>(xout, m, dn, bias, N);
    gemm_proj<<<gemmBlocks, 256, 0, stream>>>(xin, W, att_src, att_dst, h, as, ad, N);
    edge_logits_max<<<edgeBlocks, 256, 0, stream>>>(srcI, dstI, as, ad, eb, m, E);
    edge_exp<<<edgeBlocks, 256, 0, stream>>>(dstI, eb, m, dn, E);
    edge_scatter<<<scatBlocks, 256, 0, stream>>>(srcI, dstI, eb, dn, h, xout, E);

    xin = xout;
  }
}